// WrappedBlock_3882650435719
// MI455X (gfx1250) — compile-verified
//
#include <hip/hip_runtime.h>
#include <hip/hip_bf16.h>

#define BB 8
#define SS 2048
#define HH 4096
#define NPOOL 6
#define NLAYERS 32
#define NGATE 7   // N_POOL + 1

typedef float v2f __attribute__((ext_vector_type(2)));
typedef float v4f __attribute__((ext_vector_type(4)));
typedef float v8f __attribute__((ext_vector_type(8)));

// ---------------------------------------------------------------------------
// Kernel 1: gate scores via V_WMMA_F32_16X16X4_F32, softmax -> probs[8][7]
// One block, 256 threads = 8 waves. Wave w handles K in [w*512, (w+1)*512).
// A (16x4) rows = tok (M=8 padded to 16), B (4x16) cols = gate_w rows
// (N=7 padded to 16). Padding lanes load CLAMPED (valid, duplicate) rows:
// duplicate A rows only affect D[m>=8][*] and duplicate B columns only
// affect D[*][n>=7], both of which we discard — so no value masking is
// needed and the inner loop is just 2x global_load_b64 + v_wmma.
// ---------------------------------------------------------------------------
__global__ void __launch_bounds__(256)
gate_kernel(const float* __restrict__ x,
            const float* __restrict__ gate_w,
            const float* __restrict__ gate_b,
            float* __restrict__ probs) {
    __shared__ float partial[8][8][8];   // [wave][m][n]

    const int tid   = threadIdx.x;
    const int wave  = tid >> 5;
    const int lane  = tid & 31;
    const int row   = lane & 15;         // m for A-fragment, n for B-fragment
    const int kbase = (lane >> 4) << 1;  // 0 for lanes 0-15, 2 for lanes 16-31

    // Clamp rows so every load address is in-bounds (duplicates are benign).
    const int mA = (row < BB)    ? row : 0;
    const int nB = (row < NGATE) ? row : 0;

    const float* tokp = x + (size_t)mA * SS * HH + (size_t)(SS - 1) * HH;
    const float* gwp  = gate_w + (size_t)nB * HH;

    v8f c = {};
    const int k0 = wave * 512 + kbase;
    #pragma unroll 8
    for (int kk = 0; kk < 512; kk += 4) {
        const int k = k0 + kk;
        const v2f a = *(const v2f*)(tokp + k);
        const v2f b = *(const v2f*)(gwp + k);
        // D = A(16x4) * B(4x16) + C  (fp32, matches reference precision)
        c = __builtin_amdgcn_wmma_f32_16x16x4_f32(
                /*neg_a=*/false, a, /*neg_b=*/false, b,
                /*c_mod=*/(short)0, c, /*reuse_a=*/false, /*reuse_b=*/false);
    }

    // C/D layout: lanes 0..15 hold D[m=g][n=lane] in VGPR g (g=0..7).
    // We only need m<8, n<7 -> lanes 0..6 carry everything we need.
    if (lane < NGATE) {
        #pragma unroll
        for (int g = 0; g < 8; ++g) partial[wave][g][lane] = c[g];
    }
    __syncthreads();

    if (tid < BB) {
        const int m = tid;
        float s[NGATE];
        float mx = -3.402823466e38f;
        #pragma unroll
        for (int n = 0; n < NGATE; ++n) {
            float v = gate_b[n];
            #pragma unroll
            for (int w = 0; w < 8; ++w) v += partial[w][m][n];
            s[n] = v;
            mx = fmaxf(mx, v);
        }
        float sum = 0.f;
        #pragma unroll
        for (int n = 0; n < NGATE; ++n) { s[n] = __expf(s[n] - mx); sum += s[n]; }
        const float inv = 1.f / sum;
        #pragma unroll
        for (int n = 0; n < NGATE; ++n) probs[m * NGATE + n] = s[n] * inv;
    }
}

// ---------------------------------------------------------------------------
// Kernel 2: fused streaming copy (x -> out, non-temporal b128) with the
// last-token rows replaced by the gated mixture. 512 MB of traffic dominates
// (~22 us at 23.3 TB/s); NT hints keep the 512 MB stream from thrashing the
// 192 MB L2.
// ---------------------------------------------------------------------------
__global__ void __launch_bounds__(256)
combine_copy_kernel(const float* __restrict__ x,
                    const float* __restrict__ pool,
                    const int*   __restrict__ layer_idx_p,
                    const float* __restrict__ probs,
                    float*       __restrict__ out) {
    const int  L      = *layer_idx_p;
    const long total4 = (long)BB * SS * HH / 4;              // 16,777,216
    const long stride = (long)gridDim.x * blockDim.x;

    for (long i = (long)blockIdx.x * blockDim.x + threadIdx.x;
         i < total4; i += stride) {
        const long e    = i * 4;                              // float index
        const int  h    = (int)(e & (HH - 1));                // HH = 2^12
        const long rowi = e >> 12;                            // b*S + s
        const int  s    = (int)(rowi & (SS - 1));             // SS = 2^11
        const int  b    = (int)(rowi >> 11);

        if (s == SS - 1) {
            const float* pr = probs + b * NGATE;
            v4f acc = pr[NGATE - 1] * (*(const v4f*)(x + e)); // tok term
            #pragma unroll
            for (int k = 0; k < NPOOL; ++k) {
                const v4f v = *(const v4f*)(
                    pool + (((size_t)b * NPOOL + k) * NLAYERS + (size_t)L)
                               * (size_t)HH + h);
                acc += pr[k] * v;
            }
            __builtin_nontemporal_store(acc, (v4f*)(out + e));
        } else {
            const v4f v = __builtin_nontemporal_load((const v4f*)(x + e));
            __builtin_nontemporal_store(v, (v4f*)(out + e));
        }
    }
}

// ---------------------------------------------------------------------------
extern "C" void kernel_launch(void* const* d_in, const int* in_sizes, int n_in,
                              void* d_out, int out_size, void* d_ws, size_t ws_size,
                              hipStream_t stream) {
    const float* x         = (const float*)d_in[0];
    const float* gate_w    = (const float*)d_in[1];
    const float* gate_b    = (const float*)d_in[2];
    const float* pool      = (const float*)d_in[3];
    const int*   layer_idx = (const int*)d_in[4];
    float*       out       = (float*)d_out;
    float*       probs     = (float*)d_ws;  // 8*7 floats of scratch

    gate_kernel<<<1, 256, 0, stream>>>(x, gate_w, gate_b, probs);
    combine_copy_kernel<<<8192, 256, 0, stream>>>(x, pool, layer_idx, probs, out);
}